// FW_83854941487892
// MI455X (gfx1250) — compile-verified
//
#include <hip/hip_runtime.h>

// Forward warping with z-buffer (fw_cuda.forward_warping) for MI455X (gfx1250).
// Memory/atomic-bound scatter: WMMA is structurally inapplicable (no matmul).
// CDNA5-relevant choices:
//  - wave32 codegen, no lane-width assumptions
//  - non-returning global_atomic_min_u32 / global_atomic_add_u32 (results unused)
//  - b128-vectorized streaming (4 pixels/thread in every pass; W%4==0 so a
//    4-aligned quad never wraps a row -> gy constant, gx=gx0+k)
//  - 192MB L2: zbuf(33MB)+counts(33MB) stay on-chip for the atomics.
//
// Scratch layout: zbuf lives in the `valid` output slot (d_out + 3N), counts in
// the `collision` slot (d_out + 4N); finalize overwrites both with the real
// outputs. No dependence on ws_size; everything re-initialized every call.

constexpr int W_  = 3840;
constexpr int H_  = 2160;
constexpr int N_  = W_ * H_;          // 8,294,400  (divisible by 4)
constexpr int NQ_ = N_ >> 2;          // quads per plane
constexpr int N3_ = 3 * N_;           // out region end
constexpr int N4_ = 4 * N_;           // zbuf region end
constexpr int N5_ = 5 * N_;           // counts region end

// Fused vectorized init over the whole 5N output buffer:
//   [0,3N): out = 0    [3N,4N): zbuf = +inf bits    [4N,5N): counts = 0
__global__ void fw_init_kernel(uint4* __restrict__ buf) {
    int t  = blockIdx.x * blockDim.x + threadIdx.x;   // one uint4 per thread
    int i4 = t << 2;
    if (i4 >= N5_) return;
    unsigned fill = (i4 >= N3_ && i4 < N4_) ? 0x7F800000u : 0u;
    buf[t] = make_uint4(fill, fill, fill, fill);      // global_store_b128
}

// Compute the 4 destination indices for quad t (pixels i4..i4+3).
__device__ __forceinline__ void fw_flat4(const float4& fx, const float4& fy,
                                         int i4, int (&flat)[4]) {
    int gx0 = i4 % W_;                 // 4-aligned, <= W_-4: no row wrap in quad
    int gy  = i4 / W_;
    const float* fxp = (const float*)&fx;
    const float* fyp = (const float*)&fy;
#pragma unroll
    for (int k = 0; k < 4; ++k) {
        float sx = fminf(fmaxf((float)(gx0 + k) + fxp[k], 0.0f), (float)(W_ - 1));
        float sy = fminf(fmaxf((float)gy + fyp[k], 0.0f), (float)(H_ - 1));
        // jnp.round == round-half-to-even == rintf under default RNE mode
        flat[k] = (int)rintf(sy) * W_ + (int)rintf(sx);
    }
}

// Pass 1: zbuf[flat] = min(depth) via u32 atomic (depth in [0,1) => bits ordered)
__global__ void fw_zmin_kernel(const float4* __restrict__ flowx4,
                               const float4* __restrict__ flowy4,
                               const float4* __restrict__ depth4,
                               unsigned* __restrict__ zbuf) {
    int t = blockIdx.x * blockDim.x + threadIdx.x;
    if (t >= NQ_) return;
    float4 fx = flowx4[t];             // global_load_b128 x3
    float4 fy = flowy4[t];
    float4 d  = depth4[t];
    int flat[4];
    fw_flat4(fx, fy, t << 2, flat);
    const float* dp = (const float*)&d;
#pragma unroll
    for (int k = 0; k < 4; ++k) {
        atomicMin(&zbuf[flat[k]], __float_as_uint(dp[k]));  // global_atomic_min_u32
    }
}

// Pass 2: winners scatter colors; near-sources bump collision counts
__global__ void fw_resolve_kernel(const float4* __restrict__ obj4,
                                  const float4* __restrict__ flowx4,
                                  const float4* __restrict__ flowy4,
                                  const float4* __restrict__ depth4,
                                  const unsigned* __restrict__ zbuf,
                                  float* __restrict__ out,
                                  unsigned* __restrict__ counts) {
    int t = blockIdx.x * blockDim.x + threadIdx.x;
    if (t >= NQ_) return;

    float4 fx = flowx4[t];             // global_load_b128 x6
    float4 fy = flowy4[t];
    float4 d  = depth4[t];
    float4 c0 = obj4[t];
    float4 c1 = obj4[NQ_ + t];
    float4 c2 = obj4[2 * NQ_ + t];

    int flat[4];
    fw_flat4(fx, fy, t << 2, flat);

    const float* dp  = (const float*)&d;
    const float* c0p = (const float*)&c0;
    const float* c1p = (const float*)&c1;
    const float* c2p = (const float*)&c2;

#pragma unroll
    for (int k = 0; k < 4; ++k) {
        int   f = flat[k];
        float z = __uint_as_float(zbuf[f]);   // gather (L2-resident)
        if (dp[k] <= z) {  // nearest source wins (ties measure-zero)
            out[f]          = c0p[k];
            out[N_ + f]     = c1p[k];
            out[2 * N_ + f] = c2p[k];
        }
        if (dp[k] <= z + 0.2f) {
            atomicAdd(&counts[f], 1u);        // global_atomic_add_u32
        }
    }
}

// Pass 3 (vectorized x4): valid = any(channel != 0); collision = counts >= 2.
// Overwrites the zbuf/counts scratch (aliased output slots) with final floats.
__global__ void fw_finalize_kernel(const float4* __restrict__ out4,      // C0 base
                                   float4* __restrict__ valid4,          // zbuf slot
                                   float4* __restrict__ collision4) {    // counts slot
    int t = blockIdx.x * blockDim.x + threadIdx.x;    // one float4 (4 pixels)
    if (t >= NQ_) return;

    float4 c0 = out4[t];                  // global_load_b128 x3
    float4 c1 = out4[NQ_ + t];
    float4 c2 = out4[2 * NQ_ + t];

    float4 v;
    v.x = (c0.x != 0.0f || c1.x != 0.0f || c2.x != 0.0f) ? 1.0f : 0.0f;
    v.y = (c0.y != 0.0f || c1.y != 0.0f || c2.y != 0.0f) ? 1.0f : 0.0f;
    v.z = (c0.z != 0.0f || c1.z != 0.0f || c2.z != 0.0f) ? 1.0f : 0.0f;
    v.w = (c0.w != 0.0f || c1.w != 0.0f || c2.w != 0.0f) ? 1.0f : 0.0f;

    const uint4* cnt4 = (const uint4*)collision4;
    uint4 cu = cnt4[t];                   // raw u32 counts stored there
    float4 c;
    c.x = (cu.x >= 2u) ? 1.0f : 0.0f;
    c.y = (cu.y >= 2u) ? 1.0f : 0.0f;
    c.z = (cu.z >= 2u) ? 1.0f : 0.0f;
    c.w = (cu.w >= 2u) ? 1.0f : 0.0f;

    valid4[t]     = v;                    // global_store_b128 x2
    collision4[t] = c;
}

extern "C" void kernel_launch(void* const* d_in, const int* in_sizes, int n_in,
                              void* d_out, int out_size, void* d_ws, size_t ws_size,
                              hipStream_t stream) {
    const float* obj   = (const float*)d_in[0];  // [3,H,W]
    const float* flow  = (const float*)d_in[1];  // [2,H,W]
    const float* depth = (const float*)d_in[2];  // [1,H,W]

    const float4* obj4   = (const float4*)obj;
    const float4* flowx4 = (const float4*)flow;
    const float4* flowy4 = (const float4*)(flow + (size_t)N_);  // N_*4 bytes: 16B aligned
    const float4* depth4 = (const float4*)depth;

    float*    out    = (float*)d_out;                 // [3,H,W]
    float*    valid  = out + (size_t)N3_;             // [1,H,W] (zbuf scratch)
    float*    coll   = out + (size_t)N4_;             // [1,H,W] (counts scratch)
    unsigned* zbuf   = (unsigned*)valid;
    unsigned* counts = (unsigned*)coll;

    const int T = 256;  // 8 wave32s per block on CDNA5
    fw_init_kernel<<<((N5_ >> 2) + T - 1) / T, T, 0, stream>>>((uint4*)d_out);
    fw_zmin_kernel<<<(NQ_ + T - 1) / T, T, 0, stream>>>(flowx4, flowy4, depth4, zbuf);
    fw_resolve_kernel<<<(NQ_ + T - 1) / T, T, 0, stream>>>(obj4, flowx4, flowy4,
                                                           depth4, zbuf, out, counts);
    fw_finalize_kernel<<<(NQ_ + T - 1) / T, T, 0, stream>>>(
        (const float4*)out, (float4*)valid, (float4*)coll);
}